// HaarWaveletTransform_67473936220442
// MI455X (gfx1250) — compile-verified
//
#include <hip/hip_runtime.h>

typedef __attribute__((ext_vector_type(2))) float v2f;
typedef __attribute__((ext_vector_type(8))) float v8f;

#define BATCH   16
#define CHAN    64
#define H_IN    512
#define W_IN    512
#define H_OUT   256
#define W_OUT   256

// Each wave processes one output row (256 px) of one (b,c) plane.
// Per 32-px step: 2x V_WMMA_F32_16X16X4_F32 do the Haar butterfly for
// 16 px each; results merged across lane halves -> 4 full-wave 128B stores.
__global__ __launch_bounds__(256) void
HaarWaveletTransform_wmma_kernel(const float* __restrict__ x,
                                 float* __restrict__ out) {
    const int lane = threadIdx.x & 31;
    const int wave = threadIdx.x >> 5;
    const int hl   = lane >> 4;     // 0: K=0,1 (row 2h)   1: K=2,3 (row 2h+1)
    const int n    = lane & 15;     // M index for A, N index for B/D

    // plane p in [0, B*C), output row ho in [0, 256)
    const int blocksPerPlane = H_OUT / 8;                 // 32
    const int p  = blockIdx.x / blocksPerPlane;
    const int ho = (blockIdx.x % blocksPerPlane) * 8 + wave;

    // This lane's input row: 2*ho + hl (lane halves split the two source rows)
    const float* __restrict__ inRow =
        x + (size_t)p * (H_IN * W_IN) + (size_t)(2 * ho + hl) * W_IN;

    const size_t outPlaneElems = (size_t)BATCH * CHAN * H_OUT * W_OUT;
    float* __restrict__ outLL =
        out + (size_t)p * (H_OUT * W_OUT) + (size_t)ho * W_OUT;
    float* __restrict__ outLH = outLL + outPlaneElems;
    float* __restrict__ outHL = outLL + 2 * outPlaneElems;
    float* __restrict__ outHH = outLL + 3 * outPlaneElems;

    // Constant A matrix (16x4): rows 0..3 = Haar filters * 0.5, rows 4..15 = 0.
    // sign(m,k) = (-1)^(m0*k0 + m1*k1)  (separable 1D Haar signs)
    // Lane layout (32-bit A 16x4): lanes 0-15 hold K=0,1; lanes 16-31 hold K=2,3.
    const int k0 = hl * 2;
    const int k1 = k0 + 1;
    float a0 = 0.0f, a1 = 0.0f;
    if (n < 4) {
        int neg0 = ((n & 1) & (k0 & 1)) ^ (((n >> 1) & 1) & ((k0 >> 1) & 1));
        int neg1 = ((n & 1) & (k1 & 1)) ^ (((n >> 1) & 1) & ((k1 >> 1) & 1));
        a0 = neg0 ? -0.5f : 0.5f;
        a1 = neg1 ? -0.5f : 0.5f;
    }
    v2f A;
    A.x = a0;
    A.y = a1;

    const v8f zero = {};

    #pragma unroll 2
    for (int j = 0; j < W_OUT / 32; ++j) {
        const int wo0 = j * 32;

        // B operand: lane<16 -> (a,b) = x[2h, 2w:2w+2]; lane>=16 -> (c,d) = x[2h+1, 2w:2w+2]
        // 8B-aligned, contiguous 128B per half-wave, streamed once -> non-temporal.
        v2f bA = __builtin_nontemporal_load(
            (const v2f*)(inRow + 2 * (wo0 + n)));
        v2f bB = __builtin_nontemporal_load(
            (const v2f*)(inRow + 2 * (wo0 + n) + 32));

        // D rows 0..3 (lanes 0-15) = LL,LH,HL,HH for 16 pixels.
        v8f dA = __builtin_amdgcn_wmma_f32_16x16x4_f32(
            false, A, false, bA, (short)0, zero, false, false);
        v8f dB = __builtin_amdgcn_wmma_f32_16x16x4_f32(
            false, A, false, bB, (short)0, zero, false, false);

        // Merge: lanes 0-15 keep dA (pixels wo0+n), lanes 16-31 take dB from
        // lane n (pixels wo0+16+n) -> one full-wave 128B store per output.
        float tLL = __shfl(dB[0], n, 32);
        float tLH = __shfl(dB[1], n, 32);
        float tHL = __shfl(dB[2], n, 32);
        float tHH = __shfl(dB[3], n, 32);
        float vLL = hl ? tLL : dA[0];
        float vLH = hl ? tLH : dA[1];
        float vHL = hl ? tHL : dA[2];
        float vHH = hl ? tHH : dA[3];

        __builtin_nontemporal_store(vLL, outLL + wo0 + lane);
        __builtin_nontemporal_store(vLH, outLH + wo0 + lane);
        __builtin_nontemporal_store(vHL, outHL + wo0 + lane);
        __builtin_nontemporal_store(vHH, outHH + wo0 + lane);
    }
}

extern "C" void kernel_launch(void* const* d_in, const int* in_sizes, int n_in,
                              void* d_out, int out_size, void* d_ws, size_t ws_size,
                              hipStream_t stream) {
    (void)in_sizes; (void)n_in; (void)out_size; (void)d_ws; (void)ws_size;
    const float* x = (const float*)d_in[0];
    float* out = (float*)d_out;

    // 1024 planes * 32 blocks-per-plane (8 output rows per block, 1 row/wave)
    dim3 grid(BATCH * CHAN * (H_OUT / 8));
    dim3 block(256);
    HaarWaveletTransform_wmma_kernel<<<grid, block, 0, stream>>>(x, out);
}